// G_dense_24532853195159
// MI455X (gfx1250) — compile-verified
//
#include <hip/hip_runtime.h>
#include <hip/hip_bf16.h>

typedef float v2f __attribute__((ext_vector_type(2)));
typedef float v8f __attribute__((ext_vector_type(8)));

#define LEAKY 0.01f
#define BATCH 128          // M of every GEMM
#define MT    8            // 8 m-tiles of 16 = 128 rows per wave

// ---------------------------------------------------------------------------
// C[128 x N] = act( A[128 x K] @ W[K x N] + bias )
// One wave computes a 128x16 column strip (8 accumulators of 16x16 f32) using
// V_WMMA_F32_16X16X4_F32.  Software-pipelined: fragments for step k+4 are
// loaded while the 8 WMMAs for step k issue, so loads overlap matrix ops.
// Requires N % 64 == 0, K % 8 == 0 (true for all calls: N=512/1024/10240,
// K=512/1024).
// ---------------------------------------------------------------------------
__global__ void __launch_bounds__(128)
gemm_wmma_f32(const float* __restrict__ A, const float* __restrict__ W,
              const float* __restrict__ bias, float* __restrict__ C,
              int N, int K, int apply_act)
{
    const int lane = threadIdx.x & 31;
    const int wave = threadIdx.x >> 5;       // 0..3
    const int lo   = lane & 15;              // row (A) / col (B,C) index
    const int hi   = lane >> 4;              // K-pair select / row-half select
    const int n    = (blockIdx.x * 4 + wave) * 16 + lo;   // this wave's column

    v8f acc[MT];
#pragma unroll
    for (int t = 0; t < MT; ++t) acc[t] = (v8f){0.f,0.f,0.f,0.f,0.f,0.f,0.f,0.f};

    // Per-lane base pointers; all loop-carried updates are simple adds so the
    // backend strength-reduces to pointer increments + immediate offsets.
    const float* Ap = A + (size_t)lo * K + 2 * hi;      // m-tile 0, this lane's K-pair
    const float* Wp = W + (size_t)(2 * hi) * N + n;     // K-rows 2hi/2hi+1, col n
    const size_t aTile = (size_t)16 * K;                // A stride between m-tiles
    const size_t wStep = (size_t)4 * N;                 // W stride per k-step

    v2f a0[MT], a1[MT], b0, b1;

    // Fragment load: A via 8B-aligned b64 (lo*K + 2hi is even), B as two b32
    // rows k+2hi and k+2hi+1 at column n.  Advances pointers by one k-step.
#define LOAD_FRAG(AF, BF)                                        \
    do {                                                         \
        (BF).x = Wp[0];                                          \
        (BF).y = Wp[N];                                          \
        _Pragma("unroll")                                        \
        for (int mt = 0; mt < MT; ++mt)                          \
            (AF)[mt] = *(const v2f*)(Ap + mt * aTile);           \
        Wp += wStep;                                             \
        Ap += 4;                                                 \
    } while (0)

#define MMA_STEP(AF, BF)                                         \
    do {                                                         \
        _Pragma("unroll")                                        \
        for (int mt = 0; mt < MT; ++mt)                          \
            acc[mt] = __builtin_amdgcn_wmma_f32_16x16x4_f32(     \
                false, (AF)[mt], false, (BF), (short)0,          \
                acc[mt], false, false);                          \
    } while (0)

    LOAD_FRAG(a0, b0);                       // k = 0
    const int steps2 = (K >> 3) - 1;         // K/8 - 1 double-steps
    for (int s = 0; s < steps2; ++s) {
        LOAD_FRAG(a1, b1);                   // prefetch k+4
        MMA_STEP(a0, b0);                    // compute k
        LOAD_FRAG(a0, b0);                   // prefetch k+8
        MMA_STEP(a1, b1);                    // compute k+4
    }
    LOAD_FRAG(a1, b1);                       // k = K-4
    MMA_STEP(a0, b0);                        // k = K-8
    MMA_STEP(a1, b1);                        // k = K-4

#undef LOAD_FRAG
#undef MMA_STEP

    const float bv = bias ? bias[n] : 0.0f;
#pragma unroll
    for (int mt = 0; mt < MT; ++mt) {
#pragma unroll
        for (int i = 0; i < 8; ++i) {
            const int m = mt * 16 + hi * 8 + i;     // C/D layout: VGPR i -> row 8*hi+i
            float v = acc[mt][i] + bv;
            if (apply_act) v = (v >= 0.f) ? v : LEAKY * v;
            C[(size_t)m * N + n] = v;
        }
    }
}

// ---------------------------------------------------------------------------
// Minibatch discrimination.
// mm : [128][512*20] row-major (= m[b, o, k]).   One block per o (512 blocks).
// ob[j, o] = sum_i exp( -sum_k |m[i,o,k] - m[j,o,k]| ) - 1
// Stage m[:, o, :] (128*20 floats = 10 KB) in LDS, thread j sweeps all i.
// ---------------------------------------------------------------------------
#define MBD_O 512
#define MBD_K 20

__global__ void __launch_bounds__(128)
mbd_kernel(const float* __restrict__ mm, float* __restrict__ ob)
{
    __shared__ float sm[BATCH * MBD_K];
    const int o = blockIdx.x;
    const int j = threadIdx.x;               // 0..127

    for (int t = threadIdx.x; t < BATCH * MBD_K; t += blockDim.x) {
        const int b = t / MBD_K;
        const int k = t - b * MBD_K;
        sm[t] = mm[(size_t)b * (MBD_O * MBD_K) + o * MBD_K + k];
    }
    __syncthreads();

    float my[MBD_K];
#pragma unroll
    for (int k = 0; k < MBD_K; ++k) my[k] = sm[j * MBD_K + k];

    float s = 0.f;
    for (int i = 0; i < BATCH; ++i) {
        float nrm = 0.f;
#pragma unroll
        for (int k = 0; k < MBD_K; ++k) nrm += fabsf(sm[i * MBD_K + k] - my[k]);
        s += expf(-nrm);                     // i==j contributes exp(0)=1
    }
    ob[(size_t)j * MBD_O + o] = s - 1.0f;    // cancels the i==j term
}

// ---------------------------------------------------------------------------
// out[b] = [h3[b,:], ob[b,:]] @ Wc + bc      (H=1024, MBD_O=512, N_OUT=1)
// One block of 256 threads per batch row, LDS tree reduction.
// ---------------------------------------------------------------------------
#define HDIM 1024

__global__ void __launch_bounds__(256)
final_kernel(const float* __restrict__ h3, const float* __restrict__ ob,
             const float* __restrict__ Wc, const float* __restrict__ bc,
             float* __restrict__ out)
{
    __shared__ float red[256];
    const int b = blockIdx.x;
    float s = 0.f;
    for (int t = threadIdx.x; t < HDIM; t += 256)
        s += h3[(size_t)b * HDIM + t] * Wc[t];
    for (int t = threadIdx.x; t < MBD_O; t += 256)
        s += ob[(size_t)b * MBD_O + t] * Wc[HDIM + t];
    red[threadIdx.x] = s;
    __syncthreads();
    for (int off = 128; off > 0; off >>= 1) {
        if ((int)threadIdx.x < off) red[threadIdx.x] += red[threadIdx.x + off];
        __syncthreads();
    }
    if (threadIdx.x == 0) out[b] = red[0] + bc[0];
}

// ---------------------------------------------------------------------------
extern "C" void kernel_launch(void* const* d_in, const int* in_sizes, int n_in,
                              void* d_out, int out_size, void* d_ws, size_t ws_size,
                              hipStream_t stream)
{
    (void)in_sizes; (void)n_in; (void)out_size; (void)ws_size;

    const float* x  = (const float*)d_in[0];   // [128, 512]
    const float* W1 = (const float*)d_in[1];   // [512, 512]
    const float* b1 = (const float*)d_in[2];   // [512]
    const float* W2 = (const float*)d_in[3];   // [512, 1024]
    const float* b2 = (const float*)d_in[4];   // [1024]
    const float* W3 = (const float*)d_in[5];   // [1024, 1024]
    const float* b3 = (const float*)d_in[6];   // [1024]
    const float* T  = (const float*)d_in[7];   // [1024, 512, 20] == [1024, 10240]
    const float* Wc = (const float*)d_in[8];   // [1536, 1]
    const float* bc = (const float*)d_in[9];   // [1]
    float* out = (float*)d_out;                // [128, 1]

    char* ws = (char*)d_ws;                    // need ~6.6 MB
    float* h1 = (float*)(ws + 0);              // 128*512  *4 = 256 KB
    float* h2 = (float*)(ws + 262144);         // 128*1024 *4 = 512 KB
    float* h3 = (float*)(ws + 786432);         // 128*1024 *4 = 512 KB
    float* mm = (float*)(ws + 1310720);        // 128*10240*4 = 5 MB
    float* ob = (float*)(ws + 6553600);        // 128*512  *4 = 256 KB

    const dim3 blk(128);
    // layer 1: [128,512] @ [512,512]
    gemm_wmma_f32<<<dim3(512 / 64),   blk, 0, stream>>>(x,  W1, b1, h1, 512,   512,  1);
    // layer 2: [128,512] @ [512,1024]
    gemm_wmma_f32<<<dim3(1024 / 64),  blk, 0, stream>>>(h1, W2, b2, h2, 1024,  512,  1);
    // layer 3: [128,1024] @ [1024,1024]
    gemm_wmma_f32<<<dim3(1024 / 64),  blk, 0, stream>>>(h2, W3, b3, h3, 1024,  1024, 1);
    // m = h3 @ T: [128,1024] @ [1024,10240]
    gemm_wmma_f32<<<dim3(10240 / 64), blk, 0, stream>>>(h3, T, nullptr, mm, 10240, 1024, 0);
    // minibatch discrimination
    mbd_kernel<<<dim3(MBD_O), dim3(128), 0, stream>>>(mm, ob);
    // classifier
    final_kernel<<<dim3(BATCH), dim3(256), 0, stream>>>(h3, ob, Wc, bc, out);
}